// TetraSMPL_11751030522286
// MI455X (gfx1250) — compile-verified
//
#include <hip/hip_runtime.h>
#include <math.h>

typedef float v2f __attribute__((ext_vector_type(2)));
typedef float v8f __attribute__((ext_vector_type(8)));

#define NV    50000
#define NB    64
#define M3    150000          // NV*3 rows of the big GEMMs (divisible by 16)
#define MT    9375            // M3/16 row tiles
#define KS    300             // shapedirs K
#define KP    207             // posedirs K
#define KPP   208             // padded K for posedirs GEMM
#define JCH   256             // v-chunk for J reduction
#define NCH   196             // ceil(50000/256)
#define VPB   8               // vertices per block in skinning

__device__ __constant__ int c_par[23] =
    {0,0,0,1,2,3,4,5,6,7,8,9,9,9,12,13,14,16,17,18,19,20,21};

// ---------------------------------------------------------------------------
// K1: VS[vd][b] = sum_s shapedirs[vd][s]*beta[b][s] + v_template[vd]
// WMMA f32 16x16x4. One wave = one 16-row tile x all 64 cols (4 n-tiles).
// ---------------------------------------------------------------------------
__global__ __launch_bounds__(256)
void k_shape_gemm(const float* __restrict__ A,     // [150000][300]
                  const float* __restrict__ beta,  // [64][300]
                  const float* __restrict__ vt,    // [150000]
                  float* __restrict__ VS)          // [150000][64]
{
    const int wave = (int)((blockIdx.x * 256u + threadIdx.x) >> 5);
    const int lane = threadIdx.x & 31;
    if (wave >= MT) return;                    // wave-uniform: EXEC stays full
    const int mrow = lane & 15;
    const int koff = (lane >> 4) << 1;         // K offset 0 or 2 per half-wave
    const float* ap = A + (size_t)(wave * 16 + mrow) * KS + koff;

    v8f acc[4] = {};
    for (int k = 0; k < KS; k += 4) {
        v2f a = *(const v2f*)(ap + k);         // A[row][k+koff .. +1], 8B aligned
#pragma unroll
        for (int nt = 0; nt < 4; ++nt) {
            const int col = nt * 16 + mrow;
            v2f b = *(const v2f*)(beta + (size_t)col * KS + k + koff);
            acc[nt] = __builtin_amdgcn_wmma_f32_16x16x4_f32(
                false, a, false, b, (short)0, acc[nt], false, false);
        }
    }
    const int crow = wave * 16 + ((lane >> 4) << 3);   // C rows r / r+8 split
#pragma unroll
    for (int r = 0; r < 8; ++r) {
        const float tv = vt[crow + r];
#pragma unroll
        for (int nt = 0; nt < 4; ++nt)
            VS[(size_t)(crow + r) * NB + nt * 16 + mrow] = acc[nt][r] + tv;
    }
}

// ---------------------------------------------------------------------------
// K2: split-K partials of J = J_regressor @ v_shaped   (deterministic)
// block = 192 threads (t = d*64+b), one v-chunk; Jreg chunk staged in LDS.
// ---------------------------------------------------------------------------
__global__ __launch_bounds__(192)
void k_jreg(const float* __restrict__ Jreg,    // [24][50000]
            const float* __restrict__ VS,      // [150000][64]
            float* __restrict__ Jpart)         // [NCH][64*72]
{
    __shared__ float JL[24 * JCH];             // 24 KB
    const int v0 = blockIdx.x * JCH;
    const int nv = min(JCH, NV - v0);
    for (int idx = threadIdx.x; idx < 24 * JCH; idx += 192) {
        const int j = idx / JCH, vi = idx % JCH;
        JL[idx] = (vi < nv) ? Jreg[(size_t)j * NV + v0 + vi] : 0.f;
    }
    __syncthreads();

    const int t = threadIdx.x;                 // addr = v*192 + t is coalesced
    const int d = t >> 6, b = t & 63;
    float acc[24];
#pragma unroll
    for (int j = 0; j < 24; ++j) acc[j] = 0.f;
    for (int vi = 0; vi < nv; ++vi) {
        const float x = VS[(size_t)(v0 + vi) * 192 + t];
#pragma unroll
        for (int j = 0; j < 24; ++j) acc[j] += JL[j * JCH + vi] * x;
    }
    float* out = Jpart + (size_t)blockIdx.x * (NB * 72);
#pragma unroll
    for (int j = 0; j < 24; ++j) out[b * 72 + j * 3 + d] = acc[j];
}

__global__ void k_jreduce(const float* __restrict__ Jpart, float* __restrict__ J)
{
    const int e = blockIdx.x * blockDim.x + threadIdx.x;
    if (e >= NB * 72) return;
    float s = 0.f;
    for (int c = 0; c < NCH; ++c) s += Jpart[(size_t)c * (NB * 72) + e];
    J[e] = s;
}

// ---------------------------------------------------------------------------
// K3: rodrigues -> R[b][j][9] and zero-padded lrotmin[b][208]
// ---------------------------------------------------------------------------
__global__ void k_rot(const float* __restrict__ pose,
                      float* __restrict__ Rws, float* __restrict__ lrotP)
{
    const int t = blockIdx.x * blockDim.x + threadIdx.x;
    if (t >= NB * 24) return;
    const int b = t / 24, j = t % 24;
    const float x = pose[b * 72 + j * 3 + 0];
    const float y = pose[b * 72 + j * 3 + 1];
    const float z = pose[b * 72 + j * 3 + 2];
    const float ex = x + 1e-8f, ey = y + 1e-8f, ez = z + 1e-8f;
    const float ang = sqrtf(ex * ex + ey * ey + ez * ez);
    const float inv = 1.f / ang;
    const float nx = x * inv, ny = y * inv, nz = z * inv;
    const float h = 0.5f * ang;
    const float sh = sinf(h);
    float qw = cosf(h), qx = sh * nx, qy = sh * ny, qz = sh * nz;
    const float qn = 1.f / sqrtf(qw * qw + qx * qx + qy * qy + qz * qz);
    qw *= qn; qx *= qn; qy *= qn; qz *= qn;
    const float w2 = qw * qw, x2 = qx * qx, y2 = qy * qy, z2 = qz * qz;
    const float wx = qw * qx, wy = qw * qy, wz = qw * qz;
    const float xy = qx * qy, xz = qx * qz, yz = qy * qz;
    float R[9];
    R[0] = w2 + x2 - y2 - z2; R[1] = 2.f*xy - 2.f*wz;   R[2] = 2.f*wy + 2.f*xz;
    R[3] = 2.f*wz + 2.f*xy;   R[4] = w2 - x2 + y2 - z2; R[5] = 2.f*yz - 2.f*wx;
    R[6] = 2.f*xz - 2.f*wy;   R[7] = 2.f*wx + 2.f*yz;   R[8] = w2 - x2 - y2 + z2;
    float* ro = Rws + (size_t)(b * 24 + j) * 9;
#pragma unroll
    for (int k = 0; k < 9; ++k) ro[k] = R[k];
    if (j > 0) {
        float* lo = lrotP + (size_t)b * KPP + (j - 1) * 9;
#pragma unroll
        for (int k = 0; k < 9; ++k)
            lo[k] = R[k] - ((k == 0 || k == 4 || k == 8) ? 1.f : 0.f);
    } else {
        lrotP[(size_t)b * KPP + KP] = 0.f;     // zero the K pad column
    }
}

// ---------------------------------------------------------------------------
// K4: VS += posedirs @ lrotmin^T   (in place; WMMA f32 16x16x4, K padded 208)
// ---------------------------------------------------------------------------
__global__ __launch_bounds__(256)
void k_pose_gemm(const float* __restrict__ A,      // [150000][207]
                 const float* __restrict__ lrotP,  // [64][208], col 207 == 0
                 float* __restrict__ VS)           // [150000][64] (accumulate)
{
    const int wave = (int)((blockIdx.x * 256u + threadIdx.x) >> 5);
    const int lane = threadIdx.x & 31;
    if (wave >= MT) return;
    const int mrow = lane & 15;
    const int koff = (lane >> 4) << 1;
    const size_t arow = (size_t)(wave * 16 + mrow) * KP;

    v8f acc[4] = {};
    for (int k = 0; k < KPP; k += 4) {
        const int kk = k + koff;                       // <= 206 always
        v2f a;
        a.x = A[arow + kk];
        a.y = (kk < 206) ? A[arow + kk + 1] : 0.f;     // K=207 pad -> 0
#pragma unroll
        for (int nt = 0; nt < 4; ++nt) {
            const int col = nt * 16 + mrow;
            v2f b = *(const v2f*)(lrotP + (size_t)col * KPP + kk);
            acc[nt] = __builtin_amdgcn_wmma_f32_16x16x4_f32(
                false, a, false, b, (short)0, acc[nt], false, false);
        }
    }
    const int crow = wave * 16 + ((lane >> 4) << 3);
#pragma unroll
    for (int r = 0; r < 8; ++r) {
#pragma unroll
        for (int nt = 0; nt < 4; ++nt) {
            const size_t idx = (size_t)(crow + r) * NB + nt * 16 + mrow;
            VS[idx] += acc[nt][r];
        }
    }
}

// ---------------------------------------------------------------------------
// K5: kinematic chain + rest-pose correction. One thread per batch element.
// ---------------------------------------------------------------------------
__global__ void k_chain(const float* __restrict__ Rws,
                        const float* __restrict__ Jws,
                        float* __restrict__ Gws)          // [64][24][12]
{
    const int b = threadIdx.x;
    if (b >= NB) return;
    float J[24][3];
    for (int j = 0; j < 24; ++j)
        for (int d = 0; d < 3; ++d) J[j][d] = Jws[b * 72 + j * 3 + d];
    float G[24][12];
    {
        const float* R = Rws + (size_t)(b * 24) * 9;
        for (int m = 0; m < 3; ++m) {
            G[0][m*4+0] = R[m*3+0]; G[0][m*4+1] = R[m*3+1];
            G[0][m*4+2] = R[m*3+2]; G[0][m*4+3] = J[0][m];
        }
    }
    for (int i = 1; i < 24; ++i) {
        const int p = c_par[i - 1];
        const float* R = Rws + (size_t)(b * 24 + i) * 9;
        const float t0 = J[i][0] - J[p][0];
        const float t1 = J[i][1] - J[p][1];
        const float t2 = J[i][2] - J[p][2];
        for (int m = 0; m < 3; ++m) {
            const float g0 = G[p][m*4+0], g1 = G[p][m*4+1];
            const float g2 = G[p][m*4+2], g3 = G[p][m*4+3];
            G[i][m*4+0] = g0 * R[0] + g1 * R[3] + g2 * R[6];
            G[i][m*4+1] = g0 * R[1] + g1 * R[4] + g2 * R[7];
            G[i][m*4+2] = g0 * R[2] + g1 * R[5] + g2 * R[8];
            G[i][m*4+3] = g0 * t0 + g1 * t1 + g2 * t2 + g3;
        }
    }
    for (int j = 0; j < 24; ++j)
        for (int m = 0; m < 3; ++m) {
            const float corr = G[j][m*4+0] * J[j][0] + G[j][m*4+1] * J[j][1] +
                               G[j][m*4+2] * J[j][2];
            float* o = Gws + ((size_t)(b * 24) + j) * 12 + m * 4;
            o[0] = G[j][m*4+0]; o[1] = G[j][m*4+1];
            o[2] = G[j][m*4+2]; o[3] = G[j][m*4+3] - corr;
        }
}

// ---------------------------------------------------------------------------
// K6: fused skinning. block = 64 threads (one per b), VPB vertices/block.
// out[b][v][d]; per-lane float4 stores -> full cachelines.
// ---------------------------------------------------------------------------
__global__ __launch_bounds__(64)
void k_skin(const float* __restrict__ W,    // [50000][24]
            const float* __restrict__ Gws,  // [64][24][12]
            const float* __restrict__ VP,   // [150000][64] (v_posed)
            float* __restrict__ out)        // [64][50000][3]
{
    __shared__ float wsh[VPB * 24];
    const int v0 = blockIdx.x * VPB;
    const int b = threadIdx.x;
    for (int idx = b; idx < VPB * 24; idx += 64)
        wsh[idx] = W[(size_t)v0 * 24 + idx];
    __syncthreads();

    float T[VPB][12];
#pragma unroll
    for (int v = 0; v < VPB; ++v)
#pragma unroll
        for (int i = 0; i < 12; ++i) T[v][i] = 0.f;

    const float* Gb = Gws + (size_t)b * 288;
    for (int j = 0; j < 24; ++j) {
        float g[12];
#pragma unroll
        for (int i = 0; i < 12; ++i) g[i] = Gb[j * 12 + i];
#pragma unroll
        for (int v = 0; v < VPB; ++v) {
            const float w = wsh[v * 24 + j];
#pragma unroll
            for (int i = 0; i < 12; ++i) T[v][i] += w * g[i];
        }
    }
    float o[VPB * 3];
#pragma unroll
    for (int v = 0; v < VPB; ++v) {
        const size_t base = (size_t)(v0 + v) * 192 + b;   // coalesced over b
        const float p0 = VP[base], p1 = VP[base + 64], p2 = VP[base + 128];
#pragma unroll
        for (int d = 0; d < 3; ++d)
            o[v * 3 + d] = T[v][d*4+0] * p0 + T[v][d*4+1] * p1 +
                           T[v][d*4+2] * p2 + T[v][d*4+3];
    }
    float4* op = (float4*)(out + (size_t)b * (NV * 3) + (size_t)v0 * 3);
#pragma unroll
    for (int q = 0; q < (VPB * 3) / 4; ++q)
        op[q] = make_float4(o[4*q], o[4*q+1], o[4*q+2], o[4*q+3]);
}

// ---------------------------------------------------------------------------
extern "C" void kernel_launch(void* const* d_in, const int* in_sizes, int n_in,
                              void* d_out, int out_size, void* d_ws, size_t ws_size,
                              hipStream_t stream)
{
    (void)in_sizes; (void)n_in; (void)out_size; (void)ws_size;
    const float* pose      = (const float*)d_in[0];
    const float* beta      = (const float*)d_in[1];
    const float* v_templ   = (const float*)d_in[2];
    const float* shapedirs = (const float*)d_in[3];
    const float* posedirs  = (const float*)d_in[4];
    const float* Jreg      = (const float*)d_in[5];
    const float* weights   = (const float*)d_in[6];
    float* out = (float*)d_out;

    float* VS    = (float*)d_ws;                 // 9,600,000
    float* Jpart = VS    + (size_t)M3 * NB;      //   903,168
    float* Jws   = Jpart + (size_t)NCH * NB * 72;//     4,608
    float* Rws   = Jws   + NB * 72;              //    13,824
    float* lrotP = Rws   + NB * 24 * 9;          //    13,312
    float* Gws   = lrotP + NB * KPP;             //    18,432   (~42 MB total)

    const int gemm_blocks = (MT + 7) / 8;        // 8 waves per 256-thread block

    k_shape_gemm<<<gemm_blocks, 256, 0, stream>>>(shapedirs, beta, v_templ, VS);
    k_jreg<<<NCH, 192, 0, stream>>>(Jreg, VS, Jpart);
    k_jreduce<<<(NB * 72 + 255) / 256, 256, 0, stream>>>(Jpart, Jws);
    k_rot<<<(NB * 24 + 255) / 256, 256, 0, stream>>>(pose, Rws, lrotP);
    k_pose_gemm<<<gemm_blocks, 256, 0, stream>>>(posedirs, lrotP, VS);
    k_chain<<<1, 64, 0, stream>>>(Rws, Jws, Gws);
    k_skin<<<NV / VPB, 64, 0, stream>>>(weights, Gws, VS, out);
}